// Backbone_7971459301585
// MI455X (gfx1250) — compile-verified
//
#include <hip/hip_runtime.h>
#include <hip/hip_bf16.h>

typedef __attribute__((ext_vector_type(2))) float v2f;
typedef __attribute__((ext_vector_type(8))) float v8f;

#define D 64

// ---------------------------------------------------------------------------
// GEMM: C[M x 64] = A[M x 64] * W[64 x 64], fp32 via V_WMMA_F32_16X16X4_F32.
// Block = 256 threads = 8 waves. Wave w: col-tile = w&3, row-tile = 2*blk + w>>2.
// ---------------------------------------------------------------------------
__global__ __launch_bounds__(256) void gcn_gemm_wmma(
    const float* __restrict__ A, const float* __restrict__ W,
    float* __restrict__ C, int M) {
  const int lane = threadIdx.x & 31;
  const int wave = threadIdx.x >> 5;
  const int numRowTiles = (M + 15) >> 4;
  const int rowTile = blockIdx.x * 2 + (wave >> 2);
  if (rowTile >= numRowTiles) return;        // wave-uniform: EXEC stays full
  const int m0 = rowTile << 4;
  const int n0 = (wave & 3) << 4;
  const int lm = lane & 15;                  // M (or N) index within tile
  const int lh = lane >> 4;                  // K half-select (0 -> K+0/1, 1 -> K+2/3)

  v8f acc = {};
  // A fragment base: row m0+lm, K offset lh*2
  const float* arow = A + (size_t)(m0 + lm) * D + lh * 2;
  // B fragment base: col n0+lm, K offset lh*2 (W row-major [k][n], stride 64)
  const float* wcol = W + (size_t)(lh * 2) * D + n0 + lm;

#pragma unroll
  for (int k = 0; k < D; k += 4) {
    v2f a, b;
    a.x = arow[k + 0];                  // A[m0+lm][k + lh*2 + 0]
    a.y = arow[k + 1];                  // A[m0+lm][k + lh*2 + 1]
    b.x = wcol[(size_t)(k + 0) * D];    // W[k + lh*2 + 0][n0+lm]
    b.y = wcol[(size_t)(k + 1) * D];    // W[k + lh*2 + 1][n0+lm]
    acc = __builtin_amdgcn_wmma_f32_16x16x4_f32(
        /*neg_a=*/false, a, /*neg_b=*/false, b,
        /*c_mod=*/(short)0, acc, /*reuse_a=*/false, /*reuse_b=*/false);
  }

  // D layout: VGPR v -> row m0 + v + lh*8, col n0 + lm
  float* crow = C + (size_t)(m0 + lh * 8) * D + n0 + lm;
#pragma unroll
  for (int v = 0; v < 8; ++v) crow[(size_t)v * D] = acc[v];
}

// ---------------------------------------------------------------------------
// Zero a float buffer (float4 stores).
// ---------------------------------------------------------------------------
__global__ __launch_bounds__(256) void zero_f4(float4* __restrict__ p, int n4) {
  int i = blockIdx.x * blockDim.x + threadIdx.x;
  if (i < n4) p[i] = make_float4(0.f, 0.f, 0.f, 0.f);
}

// ---------------------------------------------------------------------------
// Scatter: agg[dst] += edge_w[e] * hw[src].  16 lanes per edge, float4 each
// (256B coalesced gather per edge; 4 non-returning fp32 atomics per lane).
// Working set (12.8 MB) lives in L2, so gathers+atomics resolve in L2.
// ---------------------------------------------------------------------------
__global__ __launch_bounds__(256) void gcn_scatter(
    const float* __restrict__ hw, const long long* __restrict__ ei,
    const float* __restrict__ ew, float* __restrict__ agg, int E) {
  int t = blockIdx.x * blockDim.x + threadIdx.x;
  int e = t >> 4;
  if (e >= E) return;
  int f = (t & 15) << 2;                        // feature quad
  int s = (int)ei[e];                           // src
  int d = (int)ei[(size_t)E + e];               // dst
  float w = ew[e];
  const float4 v = *(const float4*)(hw + (size_t)s * D + f);
  float* base = agg + (size_t)d * D + f;
  atomicAdd(base + 0, w * v.x);
  atomicAdd(base + 1, w * v.y);
  atomicAdd(base + 2, w * v.z);
  atomicAdd(base + 3, w * v.w);
}

// ---------------------------------------------------------------------------
// LeakyReLU elementwise: dst = x > 0 ? x : 0.01*x  (float4 granularity).
// ---------------------------------------------------------------------------
__global__ __launch_bounds__(256) void gcn_lrelu(
    const float4* __restrict__ in, float4* __restrict__ out, int n4) {
  int i = blockIdx.x * blockDim.x + threadIdx.x;
  if (i >= n4) return;
  float4 v = in[i];
  v.x = v.x > 0.f ? v.x : 0.01f * v.x;
  v.y = v.y > 0.f ? v.y : 0.01f * v.y;
  v.z = v.z > 0.f ? v.z : 0.01f * v.z;
  v.w = v.w > 0.f ? v.w : 0.01f * v.w;
  out[i] = v;
}

extern "C" void kernel_launch(void* const* d_in, const int* in_sizes, int n_in,
                              void* d_out, int out_size, void* d_ws, size_t ws_size,
                              hipStream_t stream) {
  const float* x = (const float*)d_in[0];
  const long long* ei = (const long long*)d_in[1];
  const float* ew = (const float*)d_in[2];
  const float* Ws[3] = {(const float*)d_in[3], (const float*)d_in[4],
                        (const float*)d_in[5]};

  const int N = in_sizes[0] / D;   // 50000
  const int E = in_sizes[2];       // 800000
  const size_t featBytes = (size_t)N * D * sizeof(float);

  float* hw   = (float*)d_ws;                                   // N x 64
  float* agg  = (float*)((char*)d_ws + featBytes);              // N x 64
  float* hbuf = (float*)((char*)d_ws + 2 * featBytes);          // N x 64

  const int numRowTiles = (N + 15) >> 4;
  const int gemmBlocks  = (numRowTiles + 1) >> 1;
  const int n4          = (N * D) >> 2;
  const int elemBlocks  = (n4 + 255) >> 8;
  const int scatBlocks  = ((E << 4) + 255) >> 8;

  const float* hin = x;
  for (int layer = 0; layer < 3; ++layer) {
    float* hnext = (layer == 2) ? (float*)d_out : hbuf;
    gcn_gemm_wmma<<<gemmBlocks, 256, 0, stream>>>(hin, Ws[layer], hw, N);
    zero_f4<<<elemBlocks, 256, 0, stream>>>((float4*)agg, n4);
    gcn_scatter<<<scatBlocks, 256, 0, stream>>>(hw, ei, ew, agg, E);
    gcn_lrelu<<<elemBlocks, 256, 0, stream>>>((const float4*)agg,
                                              (float4*)hnext, n4);
    hin = hnext;
  }
}